// MultiScaleTrendDirectionLoss_32804960207356
// MI455X (gfx1250) — compile-verified
//
#include <hip/hip_runtime.h>
#include <stdint.h>

// Problem constants (from reference)
#define B_ 32
#define T_ 8192
#define D_ 64
#define CHUNK 128
#define NC (T_ / CHUNK)       // 64 chunks
#define STEP 32               // timesteps per async stage
#define NSTAGE (CHUNK / STEP) // 4 stages per chunk

// alphas / decays / weights  (scan runs in the alpha-scaled domain e' = e/alpha)
#define A0 0.1f
#define A1 0.3f
#define A2 0.5f
#define DK0 0.9f
#define DK1 0.7f
#define DK2 0.5f
#define INVA0 10.0f
#define INVA1 (1.0f / 0.3f)
#define INVA2 2.0f
// final per-alpha loss scale = weight * alpha^2
#define S0 (0.5f * A0 * A0)
#define S1 (0.3f * A1 * A1)
#define S2 (0.2f * A2 * A2)

// ---------------------------------------------------------------------------
// CDNA5 async global->LDS copy (ASYNCcnt-tracked). VDST vgpr holds the LDS
// byte address, addr pair holds the 64-bit global address, "off" = no saddr.
// ---------------------------------------------------------------------------
__device__ __forceinline__ void async_b128(uint32_t lds_byte, const float* g) {
  asm volatile("global_load_async_to_lds_b128 %0, %1, off"
               :: "v"(lds_byte), "v"((uint64_t)(uintptr_t)g)
               : "memory");
}
#define WAIT_ASYNC_16() asm volatile("s_wait_asynccnt 16" ::: "memory")
#define WAIT_ASYNC_0()  asm volatile("s_wait_asynccnt 0"  ::: "memory")

// Issue one stage: copy STEP*D_ floats of pred and target (8 KB each) into
// LDS. 64 threads x 8 float4 per signal -> 16 async instructions per wave.
__device__ __forceinline__ void issue_stage(const float* pb, const float* tb,
                                            uint32_t lp, uint32_t lt, int tid) {
#pragma unroll
  for (int k = 0; k < 8; ++k) {
    const int f = tid + 64 * k;             // float4 index
    async_b128(lp + f * 16, pb + f * 4);
  }
#pragma unroll
  for (int k = 0; k < 8; ++k) {
    const int f = tid + 64 * k;
    async_b128(lt + f * 16, tb + f * 4);
  }
}

// Scaled-EMA state for the 6 series (3 alphas x {pred, target}).
struct EmaSt { float ep0, ep1, ep2, et0, et1, et2; };
struct AccSt { float a0, a1, a2; };

__device__ __forceinline__ void ema_init(float x, float y, EmaSt& e) {
  e.ep0 = x * INVA0; e.ep1 = x * INVA1; e.ep2 = x * INVA2;
  e.et0 = y * INVA0; e.et1 = y * INVA1; e.et2 = y * INVA2;
}

// N fully-unrolled scan steps (summary pass: 1 FMA per series per step).
template <int N>
__device__ __forceinline__ void ema_run(const float* xr, const float* yr,
                                        int tid, EmaSt& e) {
#pragma unroll
  for (int i = 0; i < N; ++i) {
    const float x = xr[i * D_ + tid];
    const float y = yr[i * D_ + tid];
    e.ep0 = fmaf(DK0, e.ep0, x);  e.et0 = fmaf(DK0, e.et0, y);
    e.ep1 = fmaf(DK1, e.ep1, x);  e.et1 = fmaf(DK1, e.et1, y);
    e.ep2 = fmaf(DK2, e.ep2, x);  e.et2 = fmaf(DK2, e.et2, y);
  }
}

// N fully-unrolled scan + masked-loss steps. Sign tests and squared diffs are
// alpha-scale-invariant; w_a*alpha_a^2 is folded in once at the end.
template <int N>
__device__ __forceinline__ void loss_run(const float* xr, const float* yr,
                                         int tid, EmaSt& e, AccSt& a) {
#pragma unroll
  for (int i = 0; i < N; ++i) {
    const float x = xr[i * D_ + tid];
    const float y = yr[i * D_ + tid];
    {
      const float pe = fmaf(DK0, e.ep0, x), te = fmaf(DK0, e.et0, y);
      const float pd = pe - e.ep0, td = te - e.et0;
      const bool mm = ((pd > 0.f) != (td > 0.f)) || ((pd < 0.f) != (td < 0.f));
      const float df = pe - te;
      const float sel = mm ? df : 0.f;         // cndmask + fma (not mul+sel+add)
      a.a0 = fmaf(sel, df, a.a0);
      e.ep0 = pe; e.et0 = te;
    }
    {
      const float pe = fmaf(DK1, e.ep1, x), te = fmaf(DK1, e.et1, y);
      const float pd = pe - e.ep1, td = te - e.et1;
      const bool mm = ((pd > 0.f) != (td > 0.f)) || ((pd < 0.f) != (td < 0.f));
      const float df = pe - te;
      const float sel = mm ? df : 0.f;
      a.a1 = fmaf(sel, df, a.a1);
      e.ep1 = pe; e.et1 = te;
    }
    {
      const float pe = fmaf(DK2, e.ep2, x), te = fmaf(DK2, e.et2, y);
      const float pd = pe - e.ep2, td = te - e.et2;
      const bool mm = ((pd > 0.f) != (td > 0.f)) || ((pd < 0.f) != (td < 0.f));
      const float df = pe - te;
      const float sel = mm ? df : 0.f;
      a.a2 = fmaf(sel, df, a.a2);
      e.ep2 = pe; e.et2 = te;
    }
  }
}

// ---------------------------------------------------------------------------
// Kernel A: per-chunk zero-carry scaled-EMA endpoints. S[b][c][d][s], s=a*2+sig.
// ---------------------------------------------------------------------------
__global__ __launch_bounds__(64) void kA(const float* __restrict__ pred,
                                         const float* __restrict__ targ,
                                         float* __restrict__ S) {
  __shared__ __align__(16) float shp[2][STEP * D_];
  __shared__ __align__(16) float sht[2][STEP * D_];
  const int c = blockIdx.x, b = blockIdx.y, tid = threadIdx.x;
  const size_t base = ((size_t)b * T_ + (size_t)c * CHUNK) * D_;
  const float* pb = pred + base;
  const float* tb = targ + base;
  const uint32_t lp[2] = {(uint32_t)(uintptr_t)&shp[0][0], (uint32_t)(uintptr_t)&shp[1][0]};
  const uint32_t lt[2] = {(uint32_t)(uintptr_t)&sht[0][0], (uint32_t)(uintptr_t)&sht[1][0]};

  issue_stage(pb, tb, lp[0], lt[0], tid);

  EmaSt e = {0.f, 0.f, 0.f, 0.f, 0.f, 0.f};
  for (int s = 0; s < NSTAGE; ++s) {
    if (s + 1 < NSTAGE) {
      issue_stage(pb + (s + 1) * STEP * D_, tb + (s + 1) * STEP * D_,
                  lp[(s + 1) & 1], lt[(s + 1) & 1], tid);
      WAIT_ASYNC_16();          // drain stage s, keep stage s+1 in flight
    } else {
      WAIT_ASYNC_0();
    }
    __syncthreads();
    const int buf = s & 1;
    const float* xr = &shp[buf][0];
    const float* yr = &sht[buf][0];
    if (c == 0 && s == 0) {     // ema_0 = x_0 -> e'_0 = x_0/alpha, then 31 steps
      ema_init(xr[tid], yr[tid], e);
      ema_run<STEP - 1>(xr + D_, yr + D_, tid, e);
    } else {
      ema_run<STEP>(xr, yr, tid, e);
    }
    __syncthreads();
  }
  float* o = S + ((size_t)(b * NC + c) * D_ + tid) * 6;
  o[0] = e.ep0; o[1] = e.et0; o[2] = e.ep1; o[3] = e.et1; o[4] = e.ep2; o[5] = e.et2;
}

// ---------------------------------------------------------------------------
// Kernel B: cross-chunk carry recurrence E_c = s_c + (1-a)^CHUNK * E_{c-1}
// (linear, so valid unchanged in the scaled domain).
// ---------------------------------------------------------------------------
__global__ __launch_bounds__(64) void kB(const float* __restrict__ S,
                                         float* __restrict__ E) {
  const int b = blockIdx.x, d = threadIdx.x;
  const float dec[3] = {__powf(DK0, (float)CHUNK), __powf(DK1, (float)CHUNK),
                        __powf(DK2, (float)CHUNK)};
#pragma unroll
  for (int s = 0; s < 6; ++s) {
    const float dk = dec[s >> 1];
    float e = 0.f;
    for (int c = 0; c < NC; ++c) {
      const size_t idx = ((size_t)(b * NC + c) * D_ + d) * 6 + s;
      const float sv = S[idx];
      e = (c == 0) ? sv : fmaf(dk, e, sv);
      E[idx] = e;
    }
  }
}

// ---------------------------------------------------------------------------
// Kernel C: exact scaled scan seeded with carry, masked loss, block partials.
// ---------------------------------------------------------------------------
__global__ __launch_bounds__(64) void kC(const float* __restrict__ pred,
                                         const float* __restrict__ targ,
                                         const float* __restrict__ E,
                                         float* __restrict__ P) {
  __shared__ __align__(16) float shp[2][STEP * D_];
  __shared__ __align__(16) float sht[2][STEP * D_];
  __shared__ float red[D_];
  const int c = blockIdx.x, b = blockIdx.y, tid = threadIdx.x;
  const size_t base = ((size_t)b * T_ + (size_t)c * CHUNK) * D_;
  const float* pb = pred + base;
  const float* tb = targ + base;
  const uint32_t lp[2] = {(uint32_t)(uintptr_t)&shp[0][0], (uint32_t)(uintptr_t)&shp[1][0]};
  const uint32_t lt[2] = {(uint32_t)(uintptr_t)&sht[0][0], (uint32_t)(uintptr_t)&sht[1][0]};

  EmaSt e = {0.f, 0.f, 0.f, 0.f, 0.f, 0.f};
  if (c > 0) {  // carry == scaled EMA at t0-1 == prev value for first diff
    const float* ec = E + ((size_t)(b * NC + (c - 1)) * D_ + tid) * 6;
    e.ep0 = ec[0]; e.et0 = ec[1]; e.ep1 = ec[2]; e.et1 = ec[3]; e.ep2 = ec[4]; e.et2 = ec[5];
  }
  AccSt a = {0.f, 0.f, 0.f};

  issue_stage(pb, tb, lp[0], lt[0], tid);
  for (int s = 0; s < NSTAGE; ++s) {
    if (s + 1 < NSTAGE) {
      issue_stage(pb + (s + 1) * STEP * D_, tb + (s + 1) * STEP * D_,
                  lp[(s + 1) & 1], lt[(s + 1) & 1], tid);
      WAIT_ASYNC_16();
    } else {
      WAIT_ASYNC_0();
    }
    __syncthreads();
    const int buf = s & 1;
    const float* xr = &shp[buf][0];
    const float* yr = &sht[buf][0];
    if (c == 0 && s == 0) {     // t==0: init only, no loss term; then 31 steps
      ema_init(xr[tid], yr[tid], e);
      loss_run<STEP - 1>(xr + D_, yr + D_, tid, e, a);
    } else {
      loss_run<STEP>(xr, yr, tid, e, a);
    }
    __syncthreads();
  }
  red[tid] = S0 * a.a0 + S1 * a.a1 + S2 * a.a2;   // fold w_a * alpha_a^2
  __syncthreads();
  if (tid == 0) {
    float ssum = 0.f;
    for (int i = 0; i < D_; ++i) ssum += red[i];
    P[b * NC + c] = ssum;
  }
}

// ---------------------------------------------------------------------------
// Kernel D: deterministic final reduction of B_*NC partials -> scalar mean.
// ---------------------------------------------------------------------------
__global__ __launch_bounds__(256) void kD(const float* __restrict__ P,
                                          float* __restrict__ out) {
  __shared__ float r[256];
  float s = 0.f;
  for (int i = threadIdx.x; i < B_ * NC; i += 256) s += P[i];
  r[threadIdx.x] = s;
  __syncthreads();
  for (int off = 128; off > 0; off >>= 1) {
    if (threadIdx.x < off) r[threadIdx.x] += r[threadIdx.x + off];
    __syncthreads();
  }
  if (threadIdx.x == 0)
    out[0] = r[0] * (1.0f / ((float)(T_ - 1) * (float)(B_ * D_)));
}

// ---------------------------------------------------------------------------
extern "C" void kernel_launch(void* const* d_in, const int* in_sizes, int n_in,
                              void* d_out, int out_size, void* d_ws, size_t ws_size,
                              hipStream_t stream) {
  (void)in_sizes; (void)n_in; (void)out_size; (void)ws_size;
  const float* pred = (const float*)d_in[0];
  const float* targ = (const float*)d_in[1];
  float* S = (float*)d_ws;                            // B*NC*D*6 floats (3 MB)
  float* E = S + (size_t)B_ * NC * D_ * 6;            // B*NC*D*6 floats (3 MB)
  float* P = E + (size_t)B_ * NC * D_ * 6;            // B*NC floats (8 KB)

  kA<<<dim3(NC, B_), D_, 0, stream>>>(pred, targ, S);
  kB<<<B_, D_, 0, stream>>>(S, E);
  kC<<<dim3(NC, B_), D_, 0, stream>>>(pred, targ, E, P);
  kD<<<1, 256, 0, stream>>>(P, (float*)d_out);
}